// ChaoticLogisticNet_15831249453685
// MI455X (gfx1250) — compile-verified
//
#include <hip/hip_runtime.h>

#define HIDDEN 128
#define WIDTH  512
#define ROWS_PER_BLOCK 16
#define R_MIN_C  2.6f
#define R_SPAN_C 0.6f
#define BETA_C   0.1f
#define EPS_C    1e-6f

typedef __attribute__((ext_vector_type(2))) float v2f;
typedef __attribute__((ext_vector_type(8))) float v8f;

// One transcendental per sigmoid on gfx1250 (v_tanh_f32); fallback exp2+rcp.
__device__ __forceinline__ float sigmoid_fast(float z) {
#if __has_builtin(__builtin_amdgcn_tanhf)
    return fmaf(0.5f, __builtin_amdgcn_tanhf(0.5f * z), 0.5f);
#else
    float e = __builtin_amdgcn_exp2f(z * -1.44269504088896340736f);
    return __builtin_amdgcn_rcpf(1.0f + e);
#endif
}

__global__ __launch_bounds__(512)
void chaotic_net_kernel(const float* __restrict__ x,
                        const float* __restrict__ r_w,
                        const float* __restrict__ r_b,
                        const float* __restrict__ out_w,
                        const float* __restrict__ out_b,
                        float* __restrict__ out) {
    __shared__ float lds_h[ROWS_PER_BLOCK * HIDDEN];   // 8 KB f32 tile

    const int tid  = threadIdx.x;
    const int lane = tid & 31;
    const int wv   = tid >> 5;                         // wave id == local row
    const int row  = blockIdx.x * ROWS_PER_BLOCK + wv;

    // Per-lane recurrence constants: chains j = lane + 32*k
    float rwv[4], rbv[4];
#pragma unroll
    for (int k = 0; k < 4; ++k) {
        rwv[k] = r_w[lane + 32 * k];
        rbv[k] = r_b[lane + 32 * k];
    }

    float h[4] = {0.5f, 0.5f, 0.5f, 0.5f};

    // x[row, t] is wave-uniform; read 4 steps per load.
    const float4* xrow = reinterpret_cast<const float4*>(x + (size_t)row * WIDTH);

    for (int t4 = 0; t4 < WIDTH / 4; ++t4) {
        float4 u4 = xrow[t4];
        float us[4] = {u4.x, u4.y, u4.z, u4.w};
#pragma unroll
        for (int s = 0; s < 4; ++s) {
            const float u = us[s];
#pragma unroll
            for (int k = 0; k < 4; ++k) {
                float z  = fmaf(u, rwv[k], rbv[k]);
                float sg = sigmoid_fast(z);
                float r  = fmaf(R_SPAN_C, sg, R_MIN_C);        // 2.6 + 0.6*sig
                float p  = fmaf(-h[k], h[k], h[k]);            // h*(1-h)
                float hn = fmaf(BETA_C * r, p, (1.0f - BETA_C) * h[k]);
                hn = fminf(hn, 1.0f - EPS_C);
                hn = fmaxf(hn, EPS_C);
                h[k] = hn;
            }
        }
    }

    // Stage the 16x128 f32 h-tile in LDS.
#pragma unroll
    for (int k = 0; k < 4; ++k)
        lds_h[wv * HIDDEN + lane + 32 * k] = h[k];

    __syncthreads();

    // Epilogue GEMV via V_WMMA_F32_16X16X4_F32:
    //   D(16x16) += A(16x4 of h) * B(4x16 = w[k] broadcast over columns)
    // Every wave computes it redundantly (EXEC all-1s, trivially cheap);
    // only wave 0 stores.
    // A layout (32-bit 16x4): lanes 0-15 -> {K, K+1}, lanes 16-31 -> {K+2, K+3},
    // M = lane & 15. B layout mirrors it: VGPR0 = row K/K+2, VGPR1 = row K+1/K+3.
    const int m    = lane & 15;
    const int koff = (lane >> 4) << 1;     // 0 for lanes 0-15, 2 for lanes 16-31

    v8f c = {};
#pragma unroll
    for (int kc = 0; kc < HIDDEN / 4; ++kc) {
        const int k0 = kc * 4 + koff;
        v2f a, b;
        a[0] = lds_h[m * HIDDEN + k0];
        a[1] = lds_h[m * HIDDEN + k0 + 1];
        b[0] = out_w[k0];                  // broadcast across all 16 columns
        b[1] = out_w[k0 + 1];
        c = __builtin_amdgcn_wmma_f32_16x16x4_f32(
                /*neg_a=*/false, a, /*neg_b=*/false, b,
                /*c_mod=*/(short)0, c, /*reuse_a=*/false, /*reuse_b=*/false);
    }

    // C/D layout: lane 0 holds M=0..7 in c[0..7] (N=0), lane 16 holds M=8..15.
    if (wv == 0 && (lane & 15) == 0) {
        const float ob = out_b[0];
        const int rbase = blockIdx.x * ROWS_PER_BLOCK + (lane >> 4) * 8;
#pragma unroll
        for (int i = 0; i < 8; ++i)
            out[rbase + i] = c[i] + ob;
    }
}

extern "C" void kernel_launch(void* const* d_in, const int* in_sizes, int n_in,
                              void* d_out, int out_size, void* d_ws, size_t ws_size,
                              hipStream_t stream) {
    const float* x     = (const float*)d_in[0];
    const float* r_w   = (const float*)d_in[1];
    const float* r_b   = (const float*)d_in[2];
    const float* out_w = (const float*)d_in[3];
    const float* out_b = (const float*)d_in[4];
    float* out = (float*)d_out;

    const int B = in_sizes[0] / WIDTH;              // 16384
    dim3 grid(B / ROWS_PER_BLOCK);                  // 1024 blocks
    dim3 block(512);                                // 16 waves (wave32)
    chaotic_net_kernel<<<grid, block, 0, stream>>>(x, r_w, r_b, out_w, out_b, out);

    (void)n_in; (void)out_size; (void)d_ws; (void)ws_size;
}